// projection_54597624267142
// MI455X (gfx1250) — compile-verified
//
#include <hip/hip_runtime.h>
#include <math.h>
#include <stdint.h>

// DRR cone-beam projection for MI455X (gfx1250, wave32) using the CDNA5
// Tensor Data Mover. One block = 16x16 detector tile for one pose. For each
// of the 256 sample planes, the tile's trilinear footprint is a small 3D box
// (<=24 d-rows x 2 w-slices x <=24 h) -> staged into LDS with ONE
// tensor_load_to_lds per plane (double-buffered, TENSORcnt), while waves
// compute the previous plane from LDS. Volume (64 MB) is L2-resident
// (192 MB L2), so the TDM streams L2 -> LDS off the critical path.

#define TILE 16
#define EXT  28                       // d/h extent of staged box (span<=24 + margin)
#define BUFELEM (2 * EXT * EXT)       // w(2) * d(EXT) * h(EXT) floats per buffer

typedef unsigned int u32;
typedef unsigned long long u64;
typedef __attribute__((ext_vector_type(4))) u32 u32x4;
typedef __attribute__((ext_vector_type(8))) u32 u32x8;

static __device__ __forceinline__ int iclamp(int v, int lo, int hi) {
    return v < lo ? lo : (v > hi ? hi : v);
}

struct PlaneP {
    float t;      // ray parameter for this plane
    float pw;     // uniform w coordinate
    int   w0;     // floor(pw)
    int   d_org;  // staged box origin (clamped >= 0)
    int   h_org;
    bool  ok;     // box fits in EXT -> LDS path
};

__global__ __launch_bounds__(256) void drr_tdm_kernel(
    const float* __restrict__ vol,      // D*W*H, layout [d][w][h], h contiguous
    const float* __restrict__ spacing,  // 3
    const float* __restrict__ poses,    // p*3
    const int* __restrict__ p_res_d,
    const int* __restrict__ p_res_h,
    const int* __restrict__ p_sr_d,
    const int* __restrict__ p_sr_w,
    const int* __restrict__ p_sr_h,
    float* __restrict__ out,
    int dim, int npose)
{
    extern __shared__ float smem[];     // 2 * BUFELEM floats, LDS offset 0

    const int D = dim, W = dim, H = dim;
    const int res_d = *p_res_d, res_h = *p_res_h;
    const int sr_d = *p_sr_d, sr_w = *p_sr_w, sr_h = *p_sr_h;
    const int nx = res_d * sr_d;
    const int ny = res_h * sr_h;
    const int Ws = sr_w * W;
    const int WH = W * H;

    const int ntx = (nx + TILE - 1) / TILE;
    const int nty = (ny + TILE - 1) / TILE;
    const int nTiles = npose * ntx * nty;

    const int tid = threadIdx.x;
    const int ly = tid & (TILE - 1);    // iy fastest -> coalesced
    const int lx = tid >> 4;

    const float stepx = (nx > 1) ? (float)(res_d - 1) / (float)(nx - 1) : 0.0f;
    const float stepy = (ny > 1) ? (float)(res_h - 1) / (float)(ny - 1) : 0.0f;
    const float gx_base = -0.5f * (float)res_d;
    const float gy_base = -0.5f * (float)res_h;

    const float cD = (float)(D - 1) / (float)D;
    const float cW = (float)(W - 1) / (float)W;
    const float cH = (float)(H - 1) / (float)H;
    const float pd_off = 0.5f * (float)(D - 1);
    const float ph_off = 0.5f * (float)(H - 1);
    const float P0y_step = (Ws > 1) ? (float)(W - 1) / (float)(Ws - 1) : 0.0f;

    const float sx = spacing[0], sy = spacing[1], sz = spacing[2];
    const bool dims_ok = (D >= EXT) && (H >= EXT);

    for (int tile = blockIdx.x; tile < nTiles; tile += gridDim.x) {
        const int tyi  = tile % nty;
        const int rest = tile / nty;
        const int txi  = rest % ntx;
        const int ip   = rest / ntx;

        const int ix = txi * TILE + lx;
        const int iy = tyi * TILE + ly;
        const bool valid = (ix < nx) && (iy < ny);
        const int ixc = valid ? ix : (nx - 1);
        const int iyc = valid ? iy : (ny - 1);

        // tile corner detector coords (uniform) and this thread's coords
        const int x1 = (txi * TILE + TILE - 1 < nx) ? txi * TILE + TILE - 1 : nx - 1;
        const int y1 = (tyi * TILE + TILE - 1 < ny) ? tyi * TILE + TILE - 1 : ny - 1;
        const float gxa = gx_base + stepx * (float)(txi * TILE);
        const float gxb = gx_base + stepx * (float)x1;
        const float gza = gy_base + stepy * (float)(tyi * TILE);
        const float gzb = gy_base + stepy * (float)y1;
        const float gx = gx_base + stepx * (float)ixc;
        const float gy = gy_base + stepy * (float)iyc;

        const float I0x = poses[ip * 3 + 0];
        const float I0y = poses[ip * 3 + 1];
        const float I0z = poses[ip * 3 + 2];

        const float rx = gx - I0x;
        const float ry = -I0y;
        const float rz = gy - I0z;
        const float inv_ry = 1.0f / ry;

        const float ax = rx * inv_ry * sx;
        const float ay = sy;
        const float az = rz * inv_ry * sz;
        const float dxlen = sqrtf(ax * ax + ay * ay + az * az);

        // uniform per-plane parameters + staged-box origin
        auto plane = [&](int s) -> PlaneP {
            PlaneP pp;
            const float P0y = P0y_step * (float)s;
            pp.t  = (P0y - I0y) * inv_ry;
            pp.pw = P0y * cW;
            pp.w0 = (int)floorf(pp.pw);
            const float pdA = (I0x + (gxa - I0x) * pp.t) * cD + pd_off;
            const float pdB = (I0x + (gxb - I0x) * pp.t) * cD + pd_off;
            const float phA = (I0z + (gza - I0z) * pp.t) * cH + ph_off;
            const float phB = (I0z + (gzb - I0z) * pp.t) * cH + ph_off;
            const int fdmin = (int)floorf(fminf(pdA, pdB));
            const int fdmax = (int)floorf(fmaxf(pdA, pdB));
            const int fhmin = (int)floorf(fminf(phA, phB));
            const int fhmax = (int)floorf(fmaxf(phA, phB));
            pp.d_org = iclamp(fdmin, 0, D - EXT);
            pp.h_org = iclamp(fhmin, 0, H - EXT);
            const int d_hi = iclamp(fdmax + 1, 0, D - 1);
            const int h_hi = iclamp(fhmax + 1, 0, H - 1);
            pp.ok = dims_ok && (pp.w0 >= 0) &&
                    (d_hi - pp.d_org < EXT) && (h_hi - pp.h_org < EXT);
            return pp;
        };

        // build D# (ISA ch.8 layout) and issue tensor_load_to_lds (wave 0)
        auto issue = [&](const PlaneP& pp, int bsel) {
            const u64 gaddr = (u64)(uintptr_t)vol +
                4ull * ((u64)(u32)pp.d_org * (u64)WH +
                        (u64)(u32)pp.w0 * (u64)H + (u64)(u32)pp.h_org);
            const u32 lds = (u32)(bsel * (BUFELEM * 4));
            const u32 td0 = (u32)(H - pp.h_org);     // remaining h (OOB -> 0)
            const u32 td1 = (u32)(W - pp.w0);        // remaining w slices
            const u64 s0 = (u64)H;                   // dim0->dim1 line stride
            const u64 s1 = (u64)WH;                  // dim1->dim2 plane stride
            u32x4 g0 = { 1u,                         // count=1, user mode
                         lds,
                         (u32)gaddr,
                         (u32)((gaddr >> 32) & 0x1FFFFFFu) | 0x80000000u }; // type=2
            u32x8 g1 = { 0x20000u,                               // data_size=4B
                         (td0 & 0xFFFFu) << 16,
                         (td0 >> 16) | ((td1 & 0xFFFFu) << 16),
                         (td1 >> 16) | ((u32)EXT << 16),         // tile_dim0=EXT
                         2u | ((u32)EXT << 16),                  // tile_dim1=2, tile_dim2=EXT
                         (u32)s0,
                         (u32)(s0 >> 32) | ((u32)(s1 & 0xFFFFu) << 16),
                         (u32)(s1 >> 16) };
            u32x4 g2 = { (u32)(D - pp.d_org), 0u, 0u, 0u };      // tensor_dim2
            u32x4 g3 = { 0u, 0u, 0u, 0u };
            asm volatile("tensor_load_to_lds %0, %1, %2, %3"
                         :: "s"(g0), "s"(g1), "s"(g2), "s"(g3)
                         : "memory");
        };

        float acc = 0.0f;

        PlaneP pc;
        if (Ws > 0) {
            pc = plane(0);
            if (tid < 32 && pc.ok) issue(pc, 0);
        }

        for (int s = 0; s < Ws; ++s) {
            if (tid < 32) __builtin_amdgcn_s_wait_tensorcnt(0);
            __syncthreads();                       // buf[s&1] ready; prev compute done

            PlaneP pn;
            const bool havn = (s + 1 < Ws);
            if (havn) {
                pn = plane(s + 1);
                if (tid < 32 && pn.ok) issue(pn, (s + 1) & 1);  // overlaps compute(s)
            }

            // ---- compute plane s ----
            const float px = I0x + rx * pc.t;
            const float pz = I0z + rz * pc.t;
            const float pd = px * cD + pd_off;
            const float pw = pc.pw;
            const float ph = pz * cH + ph_off;

            const float fd = floorf(pd), fh = floorf(ph);
            const int d0 = (int)fd, h0 = (int)fh;
            const int w0 = pc.w0;
            const float td = pd - fd, tw = pw - (float)w0, th = ph - fh;

            const float md0 = (d0 >= 0 && d0 < D) ? 1.0f : 0.0f;
            const float md1 = (d0 + 1 >= 0 && d0 + 1 < D) ? 1.0f : 0.0f;
            const float mw0 = (w0 >= 0 && w0 < W) ? 1.0f : 0.0f;
            const float mw1 = (w0 + 1 >= 0 && w0 + 1 < W) ? 1.0f : 0.0f;
            const float mh0 = (h0 >= 0 && h0 < H) ? 1.0f : 0.0f;
            const float mh1 = (h0 + 1 >= 0 && h0 + 1 < H) ? 1.0f : 0.0f;

            const float wd0 = (1.0f - td) * md0, wd1 = td * md1;
            const float ww0 = (1.0f - tw) * mw0, ww1 = tw * mw1;
            const float wh0 = (1.0f - th) * mh0, wh1 = th * mh1;

            float val;
            if (pc.ok) {
                // ---- LDS path: taps from staged box ----
                const float* bp = smem + ((s & 1) ? BUFELEM : 0);
                const int ld0 = iclamp(d0, 0, D - 1) - pc.d_org;
                const int ld1 = iclamp(d0 + 1, 0, D - 1) - pc.d_org;
                const int lw1 = iclamp(w0 + 1, 0, W - 1) - w0;   // 1 (0 at far edge, masked)
                const int lh0 = iclamp(h0, 0, H - 1) - pc.h_org;
                const int lh1 = iclamp(h0 + 1, 0, H - 1) - pc.h_org;
                const int r00 = (ld0 * 2) * EXT;
                const int r01 = (ld0 * 2 + lw1) * EXT;
                const int r10 = (ld1 * 2) * EXT;
                const int r11 = (ld1 * 2 + lw1) * EXT;
                const float v000 = bp[r00 + lh0], v001 = bp[r00 + lh1];
                const float v010 = bp[r01 + lh0], v011 = bp[r01 + lh1];
                const float v100 = bp[r10 + lh0], v101 = bp[r10 + lh1];
                const float v110 = bp[r11 + lh0], v111 = bp[r11 + lh1];
                val = wd0 * (ww0 * (v000 * wh0 + v001 * wh1) +
                             ww1 * (v010 * wh0 + v011 * wh1)) +
                      wd1 * (ww0 * (v100 * wh0 + v101 * wh1) +
                             ww1 * (v110 * wh0 + v111 * wh1));
            } else {
                // ---- fallback: direct global gather (float2 h-pair taps) ----
                const int dc0 = iclamp(d0, 0, D - 1);
                const int dc1 = iclamp(d0 + 1, 0, D - 1);
                const int wc0 = iclamp(w0, 0, W - 1);
                const int wc1 = iclamp(w0 + 1, 0, W - 1);
                const int hc  = iclamp(h0, 0, H - 2);
                const bool hlo = (h0 == hc);
                const float2 v00 = *(const float2*)(vol + (dc0 * WH + wc0 * H + hc));
                const float2 v01 = *(const float2*)(vol + (dc0 * WH + wc1 * H + hc));
                const float2 v10 = *(const float2*)(vol + (dc1 * WH + wc0 * H + hc));
                const float2 v11 = *(const float2*)(vol + (dc1 * WH + wc1 * H + hc));
                const float r00 = (hlo ? v00.x : v00.y) * wh0 + (hlo ? v00.y : v00.x) * wh1;
                const float r01 = (hlo ? v01.x : v01.y) * wh0 + (hlo ? v01.y : v01.x) * wh1;
                const float r10 = (hlo ? v10.x : v10.y) * wh0 + (hlo ? v10.y : v10.x) * wh1;
                const float r11 = (hlo ? v11.x : v11.y) * wh0 + (hlo ? v11.y : v11.x) * wh1;
                val = wd0 * (ww0 * r00 + ww1 * r01) + wd1 * (ww0 * r10 + ww1 * r11);
            }
            acc += val;
            pc = pn;
        }

        if (valid) out[(ip * nx + ix) * ny + iy] = acc * dxlen;

        __syncthreads();   // protect LDS before next tile's prologue TDM
    }
}

extern "C" void kernel_launch(void* const* d_in, const int* in_sizes, int n_in,
                              void* d_out, int out_size, void* d_ws, size_t ws_size,
                              hipStream_t stream) {
    const float* vol     = (const float*)d_in[0];
    const float* spacing = (const float*)d_in[1];
    const float* poses   = (const float*)d_in[2];
    const int* p_res_d = (const int*)d_in[3];
    const int* p_res_h = (const int*)d_in[4];
    const int* p_sr_d  = (const int*)d_in[5];
    const int* p_sr_w  = (const int*)d_in[6];
    const int* p_sr_h  = (const int*)d_in[7];
    float* out = (float*)d_out;

    long long n = (long long)in_sizes[0];
    int dim = (int)llroundf(cbrtf((float)n));
    while ((long long)dim * dim * dim < n) ++dim;
    while (dim > 1 && (long long)dim * dim * dim > n) --dim;

    const int npose = in_sizes[2] / 3;
    const int total = out_size;
    // grid-stride over tiles -> any grid size is correct; 2x pixel-blocks
    // comfortably covers ceil(nx/16)*ceil(ny/16)*p for 2D detectors
    int grid = ((total + 255) / 256) * 2;
    if (grid < 1) grid = 1;

    drr_tdm_kernel<<<grid, 256, 2 * BUFELEM * sizeof(float), stream>>>(
        vol, spacing, poses, p_res_d, p_res_h, p_sr_d, p_sr_w, p_sr_h,
        out, dim, npose);
}